// SimplePredictor_14250701488169
// MI455X (gfx1250) — compile-verified
//
#include <hip/hip_runtime.h>

typedef float v2f __attribute__((ext_vector_type(2)));
typedef float v4f __attribute__((ext_vector_type(4)));
typedef float v8f __attribute__((ext_vector_type(8)));

#define DIM     512
#define NCTX    65536
#define WPB     8                   // waves per block (256 threads, wave32)
#define RPW     16                  // rows per wave (one 16x16 WMMA tile)
#define NWAVES  (NCTX / RPW)        // 4096
#define NBLOCKS (NWAVES / WPB)      // 512

// Pass 1: per-wave squared distances for 16 rows via f32 WMMA Gram diagonal,
// then per-wave argmin -> workspace.
//
// K-permutation trick: dist^2 = sum_k diff[k]^2 is invariant to reordering K,
// and A/B share registers (transposed layouts), so each 16x16x4 chunk only
// needs 4 *distinct* data columns in its slots. That lets every lane fetch a
// contiguous float4 (b128): lanes 0-15 carry columns k0..k0+3 of their row,
// lanes 16-31 carry k0+4..k0+7, and the .lo/.hi float2 halves feed two WMMAs.
__global__ __launch_bounds__(256)
void nn_dist_kernel(const float* __restrict__ x, const float* __restrict__ ctx,
                    float* __restrict__ ws_d, int* __restrict__ ws_i)
{
    __shared__ __align__(16) float sx[DIM];
    __shared__ float diag[WPB][8][32];

    for (int t = threadIdx.x; t < DIM; t += 256) sx[t] = x[t];
    __syncthreads();

    const int lane  = threadIdx.x & 31;
    const int wav   = threadIdx.x >> 5;
    const int gwave = blockIdx.x * WPB + wav;
    const int row0  = gwave * RPW;
    const int r     = lane & 15;          // matrix row handled by this lane
    const int hi    = lane >> 4;          // half-wave select
    const int coff  = hi << 2;            // column offset 0 or 4

    const float* rowp = ctx + (size_t)(row0 + r) * DIM + coff;

    // 4 independent accumulators to pipeline the XDL C->D chain.
    v8f acc0 = {}, acc1 = {}, acc2 = {}, acc3 = {};

    for (int k0 = 0; k0 < DIM; k0 += 32) {
        v4f c0 = *(const v4f*)(rowp + k0);
        v4f c1 = *(const v4f*)(rowp + k0 + 8);
        v4f c2 = *(const v4f*)(rowp + k0 + 16);
        v4f c3 = *(const v4f*)(rowp + k0 + 24);
        v4f b0 = *(const v4f*)(&sx[k0 + coff]);
        v4f b1 = *(const v4f*)(&sx[k0 + 8 + coff]);
        v4f b2 = *(const v4f*)(&sx[k0 + 16 + coff]);
        v4f b3 = *(const v4f*)(&sx[k0 + 24 + coff]);
        v4f e0 = c0 - b0, e1 = c1 - b1, e2 = c2 - b2, e3 = c3 - b3;
        // Same regs as A (16x4) and B (4x16): D = diff * diff^T, diagonal
        // accumulates squared distances. Two chunks per float4.
        acc0 = __builtin_amdgcn_wmma_f32_16x16x4_f32(false, e0.lo, false, e0.lo, (short)0, acc0, false, false);
        acc1 = __builtin_amdgcn_wmma_f32_16x16x4_f32(false, e0.hi, false, e0.hi, (short)0, acc1, false, false);
        acc2 = __builtin_amdgcn_wmma_f32_16x16x4_f32(false, e1.lo, false, e1.lo, (short)0, acc2, false, false);
        acc3 = __builtin_amdgcn_wmma_f32_16x16x4_f32(false, e1.hi, false, e1.hi, (short)0, acc3, false, false);
        acc0 = __builtin_amdgcn_wmma_f32_16x16x4_f32(false, e2.lo, false, e2.lo, (short)0, acc0, false, false);
        acc1 = __builtin_amdgcn_wmma_f32_16x16x4_f32(false, e2.hi, false, e2.hi, (short)0, acc1, false, false);
        acc2 = __builtin_amdgcn_wmma_f32_16x16x4_f32(false, e3.lo, false, e3.lo, (short)0, acc2, false, false);
        acc3 = __builtin_amdgcn_wmma_f32_16x16x4_f32(false, e3.hi, false, e3.hi, (short)0, acc3, false, false);
    }
    v8f acc = (acc0 + acc1) + (acc2 + acc3);

    // Stage the 8 C/D VGPRs through LDS to pull out the diagonal:
    // D[m][m] sits at lane m, vgpr m (m<8) or lane m+16, vgpr m-8 (m>=8).
    #pragma unroll
    for (int i = 0; i < 8; ++i) diag[wav][i][lane] = acc[i];
    __syncthreads();

    float d = 3.402823466e38f;
    int   idx = 0x7fffffff;
    if (lane < RPW) {
        const int m = lane;
        d   = diag[wav][m & 7][(m < 8) ? m : (m + 16)];
        idx = row0 + m;
    }
    // argmin across the 16 valid lanes (tie-break: lowest index, like argmin)
    for (int off = 8; off > 0; off >>= 1) {
        float od = __shfl_down(d, off, 16);
        int   oi = __shfl_down(idx, off, 16);
        if (od < d || (od == d && oi < idx)) { d = od; idx = oi; }
    }
    if (lane == 0) { ws_d[gwave] = d; ws_i[gwave] = idx; }
}

// Pass 2: global argmin over 4096 per-wave candidates, then gather the
// winning 512-float context row into d_out.
__global__ __launch_bounds__(512)
void nn_argmin_gather(const float* __restrict__ ws_d, const int* __restrict__ ws_i,
                      const float* __restrict__ ctx, float* __restrict__ out)
{
    __shared__ float sd[512];
    __shared__ int   si[512];

    float d = 3.402823466e38f;
    int   idx = 0x7fffffff;
    for (int t = threadIdx.x; t < NWAVES; t += 512) {
        float od = ws_d[t];
        int   oi = ws_i[t];
        if (od < d || (od == d && oi < idx)) { d = od; idx = oi; }
    }
    sd[threadIdx.x] = d;
    si[threadIdx.x] = idx;
    __syncthreads();

    for (int s = 256; s > 0; s >>= 1) {
        if ((int)threadIdx.x < s) {
            float od = sd[threadIdx.x + s];
            int   oi = si[threadIdx.x + s];
            if (od < sd[threadIdx.x] || (od == sd[threadIdx.x] && oi < si[threadIdx.x])) {
                sd[threadIdx.x] = od;
                si[threadIdx.x] = oi;
            }
        }
        __syncthreads();
    }

    const int best = si[0];
    out[threadIdx.x] = ctx[(size_t)best * DIM + threadIdx.x]; // blockDim == DIM
}

extern "C" void kernel_launch(void* const* d_in, const int* in_sizes, int n_in,
                              void* d_out, int out_size, void* d_ws, size_t ws_size,
                              hipStream_t stream) {
    const float* x        = (const float*)d_in[0];   // [512]
    const float* contexts = (const float*)d_in[1];   // [65536, 512]
    float*       out      = (float*)d_out;           // [512]

    float* ws_d = (float*)d_ws;                                  // 4096 floats
    int*   ws_i = (int*)((char*)d_ws + NWAVES * sizeof(float));  // 4096 ints

    nn_dist_kernel<<<NBLOCKS, 256, 0, stream>>>(x, contexts, ws_d, ws_i);
    nn_argmin_gather<<<1, 512, 0, stream>>>(ws_d, ws_i, contexts, out);
}